// LearnedPooler2d_37340445671672
// MI455X (gfx1250) — compile-verified
//
#include <hip/hip_runtime.h>
#include <stdint.h>

// LearnedPooler2d on gfx1250 (MI455X), compile-only tuning.
// Shapes: h[B=32][N=4096][C=256] f32, mask[B][N] bool(1B), queries[Q=4][C=256] f32.
// Bandwidth-bound (134MB of h @ 23.3TB/s ~= 5.8us floor) -> single fused pass,
// split-K over positions, V_WMMA_F32_16X16X4_F32 for both GEMM stages,
// GLOBAL_LOAD_ASYNC_TO_LDS_B128 staging, and an LDS layout whose row stride
// (132 dwords < 256) lets ds_load_2addr_b32 fuse stage-2 operand pairs directly.

typedef __attribute__((ext_vector_type(2))) float v2f;  // A/B frag of 16x16x4 f32 WMMA
typedef __attribute__((ext_vector_type(8))) float v8f;  // C/D frag (16x16 f32)

#define CST   132          // LDS row stride (dwords) per 128-channel half; 132%64==4
#define HHALF (32 * CST)   // hs half size (dwords)
#define QHALF (16 * CST)   // qs half size (dwords)
#define WLDA  20           // padded stride for 16-wide weight tile
#define NPOS  4096
#define CCH   256

#if __has_builtin(__builtin_amdgcn_wmma_f32_16x16x4_f32)
__device__ __forceinline__ v8f wmma4(v2f a, v2f b, v8f c) {
  // D = A(16x4,f32) * B(4x16,f32) + C(16x16,f32)
  return __builtin_amdgcn_wmma_f32_16x16x4_f32(false, a, false, b, (short)0, c, false, false);
}
#else
#if defined(__HIP_DEVICE_COMPILE__)
#warning "DEVICE PASS: __builtin_amdgcn_wmma_f32_16x16x4_f32 not available — placeholder emitted, no v_wmma!"
#endif
__device__ __forceinline__ v8f wmma4(v2f a, v2f b, v8f c) { c[0] += a[0] * b[0]; return c; }
#endif

__device__ __forceinline__ void wait_asynccnt0() {
#if __has_builtin(__builtin_amdgcn_s_wait_asynccnt)
  __builtin_amdgcn_s_wait_asynccnt(0);
#else
  asm volatile("s_wait_asynccnt 0x0" ::: "memory");
#endif
}

// Async global->LDS copy of 16 bytes per lane (GVS mode: uniform SGPR base +
// per-lane 32-bit byte offset). Tracked by ASYNCcnt.
__device__ __forceinline__ void async_ld_lds_b128(uint32_t lds_byte_addr,
                                                  uint32_t glb_byte_off,
                                                  uint64_t glb_base) {
  asm volatile("global_load_async_to_lds_b128 %0, %1, %2"
               :: "v"(lds_byte_addr), "v"(glb_byte_off), "s"(glb_base)
               : "memory");
}

// One workgroup = 64 threads = 2 waves. Each block handles batch b, slice sl of
// NS = 4096/S positions; each wave owns 16 positions of every 32-position chunk.
// Produces partial (sum_n w*h) [4x256] and (sum_n w) [4] per wave into d_ws.
__global__ __launch_bounds__(64) void pool_partial(const float* __restrict__ h,
                                                   const unsigned char* __restrict__ mask,
                                                   const float* __restrict__ queries,
                                                   float* __restrict__ acc_ws,
                                                   float* __restrict__ l_ws,
                                                   int S, int NS) {
  __shared__ float qs[2 * QHALF];       // queries zero-padded to 16 rows, split halves
  __shared__ float hs[2 * HHALF];       // h chunk: 32 positions x (2 x 128ch halves)
  __shared__ float wtr[2 * 16 * WLDA];  // per-wave weight tile, transposed to A layout

  const int tid  = threadIdx.x;
  const int wv   = tid >> 5;
  const int lane = tid & 31;
  const int l15  = lane & 15;
  const int sel  = (lane >> 4) << 1;    // K-pair select: lanes 0-15 -> K+0/1, 16-31 -> K+2/3
  const int b    = blockIdx.x / S;
  const int sl   = blockIdx.x - b * S;
  const int n0s  = sl * NS;
  const int posB = wv * 16;             // this wave's position base inside the chunk

  // LDS aperture: flat addr low 32 bits carry the LDS byte offset (ISA 10.2).
  const uint32_t hsLds = (uint32_t)(uintptr_t)(&hs[0]);

  // Stage queries (rows 4..15 zero) and zero the weight-transpose tiles once.
  for (int j = tid; j < 16 * CCH; j += 64) {
    int r = j >> 8, c = j & 255;
    qs[(c >> 7) * QHALF + r * CST + (c & 127)] = (r < 4) ? queries[r * CCH + c] : 0.0f;
  }
  for (int j = tid; j < 2 * 16 * WLDA; j += 64) wtr[j] = 0.0f;

  const float scale = 0.0625f;  // 256^-0.5
  v8f acc[16];                  // pooled accumulator: 16 channel tiles of 16x16
  const v8f vzero = {};
#pragma unroll
  for (int t = 0; t < 16; ++t) acc[t] = vzero;
  float lsum[4] = {0.0f, 0.0f, 0.0f, 0.0f};

  const int nChunks = NS >> 5;
  for (int ch = 0; ch < nChunks; ++ch) {
    const int n0 = n0s + (ch << 5);
    __syncthreads();  // previous chunk's consumers of hs are done

    // ---- async-stage 32x256 f32 chunk of h into LDS (b128 per lane) ----
    const uint64_t srcBase = (uint64_t)(uintptr_t)(h + ((size_t)b * NPOS + n0) * CCH);
#pragma unroll 4
    for (int j = tid; j < 32 * 64; j += 64) {  // float4 units; 32 issues/lane
      int r = j >> 6, c4 = (j & 63) << 2;
      uint32_t ldsOff = (uint32_t)(((c4 >> 7) * HHALF + r * CST + (c4 & 127)) * 4);
      async_ld_lds_b128(hsLds + ldsOff, (uint32_t)((r * CCH + c4) * 4), srcBase);
    }
    wait_asynccnt0();   // this wave's copies landed in LDS
    __syncthreads();    // ... and everyone else's too

    // ---- stage 1: scores S[q][n] = sum_c Q[q][c] * h[n][c], K=256 in steps of 4 ----
    v8f Sv = vzero;
#pragma unroll 8
    for (int k = 0; k < CCH; k += 4) {
      const int hf = k >> 7, kl = (k & 127) + sel;   // never crosses a half (k%4==0)
      v2f a  = *(const v2f*)&qs[hf * QHALF + l15 * CST + kl];          // A[M=l15][k..k+1]
      v2f bb = *(const v2f*)&hs[hf * HHALF + (posB + l15) * CST + kl]; // B[k..][N=l15]
      Sv = wmma4(a, bb, Sv);
    }

    // ---- mask + exp; accumulate l; park w (rows q=0..3) in LDS in A layout ----
    float w0 = 0.f, w1 = 0.f, w2 = 0.f, w3 = 0.f;
    if (lane < 16) {  // lanes 0-15 hold rows M=0..7 -> q=0..3 live in Sv[0..3]
      const unsigned char m = mask[(size_t)b * NPOS + n0 + posB + l15];
      if (!m) {
        w0 = __expf(Sv[0] * scale);
        w1 = __expf(Sv[1] * scale);
        w2 = __expf(Sv[2] * scale);
        w3 = __expf(Sv[3] * scale);
      }
      lsum[0] += w0; lsum[1] += w1; lsum[2] += w2; lsum[3] += w3;
      float* wt = &wtr[wv * 16 * WLDA];
      wt[0 * WLDA + l15] = w0;   // wtr[q][n]; rows 4..15 stay zero
      wt[1 * WLDA + l15] = w1;
      wt[2 * WLDA + l15] = w2;
      wt[3 * WLDA + l15] = w3;
    }
    __syncthreads();

    // ---- stage 2: acc[q][c] += sum_n w[q][n] * h[n][c], K=16 positions ----
    v2f afr[4];
#pragma unroll
    for (int kk = 0; kk < 4; ++kk)  // A frags reused across all 16 channel tiles
      afr[kk] = *(const v2f*)&wtr[(wv * 16 + l15) * WLDA + (kk << 2) + sel];
#pragma unroll
    for (int t = 0; t < 16; ++t) {
      const int c0 = t << 4;
      const int hf = c0 >> 7, lc = c0 & 127;
#pragma unroll
      for (int kk = 0; kk < 4; ++kk) {
        const int row = posB + (kk << 2) + sel;     // position index = K (even)
        // Pair is CST(=132<256) dwords apart -> fusable as one ds_load_2addr_b32
        // landing directly as the WMMA operand pair (no repack movs).
        const float* pr = &hs[hf * HHALF + row * CST + lc + l15];
        v2f bb;
        bb.x = pr[0];        // B[k]  [N=c] = h[row]  [c]
        bb.y = pr[CST];      // B[k+1][N=c] = h[row+1][c]
        acc[t] = wmma4(afr[kk], bb, acc[t]);
      }
    }
  }

  // ---- write per-wave partials (fixed layout -> deterministic combine) ----
  const int p = blockIdx.x * 2 + wv;  // global partial index = b*(2S) + (sl*2+wv)
  if (lane < 16) {
#pragma unroll
    for (int t = 0; t < 16; ++t)
#pragma unroll
      for (int q = 0; q < 4; ++q)
        acc_ws[(size_t)p * 1024 + q * 256 + (t << 4) + l15] = acc[t][q];
  }
#pragma unroll
  for (int q = 0; q < 4; ++q) {
    float v = (lane < 16) ? lsum[q] : 0.0f;
    for (int off = 16; off; off >>= 1) v += __shfl_xor(v, off, 32);
    if (lane == 0) l_ws[p * 4 + q] = v;
  }
}

// out[b][q][c] = (sum_p acc_p) / (sum_p l_p), 0 if fully masked. Fixed-order sum.
__global__ __launch_bounds__(256) void pool_combine(const float* __restrict__ acc_ws,
                                                    const float* __restrict__ l_ws,
                                                    float* __restrict__ out, int P) {
  const int idx = blockIdx.x * blockDim.x + threadIdx.x;  // < 32*1024
  const int b  = idx >> 10;
  const int qc = idx & 1023;
  const int q  = qc >> 8;
  float a = 0.0f, L = 0.0f;
  for (int p = 0; p < P; ++p) {
    a += acc_ws[((size_t)b * P + p) * 1024 + qc];
    L += l_ws[((size_t)b * P + p) * 4 + q];
  }
  out[idx] = (L > 0.0f) ? (a / L) : 0.0f;
}

extern "C" void kernel_launch(void* const* d_in, const int* in_sizes, int n_in,
                              void* d_out, int out_size, void* d_ws, size_t ws_size,
                              hipStream_t stream) {
  const float*         h       = (const float*)d_in[0];
  const unsigned char* mask    = (const unsigned char*)d_in[1];  // jnp.bool_: 1 byte/elem
  const float*         queries = (const float*)d_in[2];
  float*               out     = (float*)d_out;

  // Pick largest split factor whose partials fit in d_ws (deterministic per run).
  int S = 16;
  while (S > 1 && (size_t)32 * S * 2 * 1028 * sizeof(float) > ws_size) S >>= 1;
  const int NS = NPOS / S;

  float* acc_ws = (float*)d_ws;                       // [32*S*2][4][256]
  float* l_ws   = acc_ws + (size_t)32 * S * 2 * 1024; // [32*S*2][4]

  hipLaunchKernelGGL(pool_partial, dim3(32 * S), dim3(64), 0, stream,
                     h, mask, queries, acc_ws, l_ws, S, NS);
  hipLaunchKernelGGL(pool_combine, dim3(128), dim3(256), 0, stream,
                     acc_ws, l_ws, out, 2 * S);
}